// Generator_3736621548191
// MI455X (gfx1250) — compile-verified
//
#include <hip/hip_runtime.h>
#include <math.h>

#define N_SEQ  512
#define N_POOL 4096
#define N_FEAT 20
#define E_DIM  64
#define G_DIM  128
#define S_DIM  16
#define ALPHA  20

typedef float v2f __attribute__((ext_vector_type(2)));
typedef float v8f __attribute__((ext_vector_type(8)));

// ---------------------------------------------------------------------------
// Generic f32 WMMA GEMM: C(MxN) = A(MxK) * B(KxN), row-major, N % 64 == 0.
// Each wave computes a 16x64 C stripe (4 WMMA accumulators along N), so one
// A-fragment b64 load feeds four V_WMMA_F32_16X16X4_F32 ops.
// VGPR layouts per CDNA5 ISA 7.12.2:
//   A 16x4: lane<16 holds M=lane,K={0,1}; lane>=16 holds M=lane-16,K={2,3}
//   B 4x16: lane<16 holds N=lane,K={0,1}; lane>=16 holds N=lane-16,K={2,3}
//   C/D:    VGPR r -> row m0 + r + 8*(lane>>4), col n + (lane&15)
// ---------------------------------------------------------------------------
__global__ void gemm_wmma_f32(const float* __restrict__ A,
                              const float* __restrict__ B,
                              float* __restrict__ C,
                              int M, int N, int K, int lda, int ldb, int ldc)
{
    const int wave = threadIdx.x >> 5;
    const int lane = threadIdx.x & 31;
    const int nstripes = N >> 6;                       // N / 64
    const int stripes = (M >> 4) * nstripes;
    const int s = blockIdx.x * (blockDim.x >> 5) + wave;
    if (s >= stripes) return;                          // uniform per wave -> EXEC all-1s
    const int m0 = (s / nstripes) << 4;
    const int n0 = (s % nstripes) << 6;
    const int ml = lane & 15;
    const int kh = lane >> 4;                          // K-half (0/1)

    v8f acc0 = {0.f,0.f,0.f,0.f,0.f,0.f,0.f,0.f};
    v8f acc1 = acc0, acc2 = acc0, acc3 = acc0;

    const float* ap  = A + (size_t)(m0 + ml) * lda + 2 * kh;   // A[m][k+2kh]
    const float* bp0 = B + (size_t)(2 * kh) * ldb + n0 + ml;   // B[k+2kh][n]
    const float* bp1 = bp0 + ldb;                              // B[k+2kh+1][n]

#pragma unroll 4
    for (int k = 0; k < K; k += 4) {
        v2f a;  a.x = ap[0];   a.y = ap[1];
        v2f b0; b0.x = bp0[0];  b0.y = bp1[0];
        v2f b1; b1.x = bp0[16]; b1.y = bp1[16];
        v2f b2; b2.x = bp0[32]; b2.y = bp1[32];
        v2f b3; b3.x = bp0[48]; b3.y = bp1[48];
        acc0 = __builtin_amdgcn_wmma_f32_16x16x4_f32(false, a, false, b0, (short)0, acc0, false, false);
        acc1 = __builtin_amdgcn_wmma_f32_16x16x4_f32(false, a, false, b1, (short)0, acc1, false, false);
        acc2 = __builtin_amdgcn_wmma_f32_16x16x4_f32(false, a, false, b2, (short)0, acc2, false, false);
        acc3 = __builtin_amdgcn_wmma_f32_16x16x4_f32(false, a, false, b3, (short)0, acc3, false, false);
        ap  += 4;
        bp0 += 4 * ldb;
        bp1 += 4 * ldb;
    }

    float* crow = C + (size_t)(m0 + 8 * kh) * ldc + n0 + ml;
#pragma unroll
    for (int r = 0; r < 8; ++r) {
        crow[(size_t)r * ldc +  0] = acc0[r];
        crow[(size_t)r * ldc + 16] = acc1[r];
        crow[(size_t)r * ldc + 32] = acc2[r];
        crow[(size_t)r * ldc + 48] = acc3[r];
    }
}

// ---------------------------------------------------------------------------
// f_nodes_in = f @ w_feat + b_feat ; g = relu(f_nodes_in).  grid=4096, block=64
// ---------------------------------------------------------------------------
__global__ void feat_kernel(const float* __restrict__ f,
                            const float* __restrict__ w_feat,
                            const float* __restrict__ b_feat,
                            float* __restrict__ fn, float* __restrict__ g)
{
    const int i = blockIdx.x;
    const int c = threadIdx.x;
    float acc = b_feat[c];
#pragma unroll
    for (int q = 0; q < N_FEAT; ++q)
        acc += f[i * N_FEAT + q] * w_feat[q * E_DIM + c];
    fn[(size_t)i * E_DIM + c] = acc;
    g[(size_t)i * E_DIM + c] = fmaxf(acc, 0.f);
}

// ---------------------------------------------------------------------------
// Cseq[k] = sum_c relu((x @ w_seq + b_seq)[k,c]) * w_edge[128+c]
// ---------------------------------------------------------------------------
__global__ void cseq_kernel(const float* __restrict__ x,
                            const float* __restrict__ w_seq,
                            const float* __restrict__ b_seq,
                            const float* __restrict__ w_edge,
                            float* __restrict__ Cseq)
{
    const int k = blockIdx.x * blockDim.x + threadIdx.x;
    if (k >= N_SEQ) return;
    float cs = 0.f;
    for (int c = 0; c < S_DIM; ++c) {
        float a = b_seq[c];
#pragma unroll
        for (int q = 0; q < ALPHA; ++q)
            a += x[k * ALPHA + q] * w_seq[q * S_DIM + c];
        cs += fmaxf(a, 0.f) * w_edge[2 * E_DIM + c];
    }
    Cseq[k] = cs;
}

// ---------------------------------------------------------------------------
// W2G = w_gcn2 @ w_graph (64x128), W2T = w_gcn2 @ w_gate.  grid=64, block=256
// ---------------------------------------------------------------------------
__global__ void w2_kernel(const float* __restrict__ w_gcn2,
                          const float* __restrict__ w_graph,
                          const float* __restrict__ w_gate,
                          float* __restrict__ W2G, float* __restrict__ W2T)
{
    const int e = blockIdx.x;
    const int t = threadIdx.x;
    const float* wsrc = (t < G_DIM) ? w_graph : w_gate;
    const int gcol = (t < G_DIM) ? t : (t - G_DIM);
    float a = 0.f;
#pragma unroll
    for (int c = 0; c < E_DIM; ++c)
        a += w_gcn2[e * E_DIM + c] * wsrc[c * G_DIM + gcol];
    if (t < G_DIM) W2G[e * G_DIM + gcol] = a;
    else           W2T[e * G_DIM + gcol] = a;
}

// ---------------------------------------------------------------------------
// s0[i] = sum_c relu((t3 @ w_gcn1)[i,c]) * w_edge[c].   grid=4096, block=64
// ---------------------------------------------------------------------------
__global__ void post_kernel(const float* __restrict__ t3,
                            const float* __restrict__ w_gcn1,
                            const float* __restrict__ w_edge,
                            float* __restrict__ s0)
{
    __shared__ float row[E_DIM];
    __shared__ float red[E_DIM];
    const int i = blockIdx.x;
    const int c = threadIdx.x;
    row[c] = t3[(size_t)i * E_DIM + c];
    __syncthreads();
    float h = 0.f;
#pragma unroll
    for (int e = 0; e < E_DIM; ++e)
        h += row[e] * w_gcn1[e * E_DIM + c];
    red[c] = fmaxf(h, 0.f) * w_edge[c];
    __syncthreads();
    for (int s = 32; s > 0; s >>= 1) {
        if (c < s) red[c] += red[c + s];
        __syncthreads();
    }
    if (c == 0) s0[i] = red[0];
}

// Dh[k][j] coefficient (j <= k assumed)
__device__ __forceinline__ float dhc(int j, int k)
{
    if (k == 0) return 1.0f;
    return (j == 0 || j == k) ? 0.70710678118654752f : 0.57735026918962576f;
}

__device__ __forceinline__ float wave_sum32(float v)
{
#pragma unroll
    for (int m = 16; m > 0; m >>= 1)
        v += __shfl_xor(v, m, 32);
    return v;
}

// argmax combine with first-index tie-break, within a wave32
__device__ __forceinline__ void wave_argmax32(float& bv, int& bi)
{
#pragma unroll
    for (int m = 16; m > 0; m >>= 1) {
        const float ov = __shfl_xor(bv, m, 32);
        const int   oi = __shfl_xor(bi, m, 32);
        if (ov > bv || (ov == bv && oi < bi)) { bv = ov; bi = oi; }
    }
}

// ---------------------------------------------------------------------------
// Persistent sequential scan: 1 block, 1024 threads (32 waves), 512 steps.
// ---------------------------------------------------------------------------
__global__ void __launch_bounds__(1024)
scan_kernel(const float* __restrict__ s0,
            const float* __restrict__ Cseq,
            const float* __restrict__ FW,   // f_nodes_in @ w_gcn2           (4096x64)
            const float* __restrict__ GW,   // f_nodes_in @ (w_gcn2@w_graph) (4096x128)
            const float* __restrict__ GG,   // f_nodes_in @ (w_gcn2@w_gate)  (4096x128)
            const float* __restrict__ w_edge,
            const float* __restrict__ b_edge,
            const float* __restrict__ b_graph,
            const float* __restrict__ b_gate,
            float* __restrict__ scores,
            int* __restrict__ idxs_out)
{
    __shared__ float maskS[N_POOL];
    __shared__ int   idxS[N_SEQ];
    __shared__ float hlast[E_DIM];
    __shared__ float hgraph[G_DIM];
    __shared__ float redv[32];
    __shared__ int   redi[32];
    __shared__ float part[32 * G_DIM];
    __shared__ float ckS;

    const int t    = threadIdx.x;
    const int lane = t & 31;
    const int wv   = t >> 5;

    for (int n = t; n < N_POOL; n += 1024) maskS[n] = 1.0f;
    if (t < E_DIM) hlast[t] = 0.0f;
    if (t < G_DIM) hgraph[t] = 0.0f;
    const float be0 = b_edge[0];

    const int cg = (t & 31) << 2;   // column base for h_graph partition (0..124)
    const int jg = t >> 5;          // row partition 0..31
    const float bg0 = b_graph[cg + 0], bg1 = b_graph[cg + 1];
    const float bg2 = b_graph[cg + 2], bg3 = b_graph[cg + 3];
    const float bt0 = b_gate[cg + 0], bt1 = b_gate[cg + 1];
    const float bt2 = b_gate[cg + 2], bt3 = b_gate[cg + 3];
    __syncthreads();

    for (int k = 0; k < N_SEQ; ++k) {
        // ---- C_k = relu(hlast).we[64:128] + relu(hgraph).we[144:272] + Cseq[k]
        float c = 0.f;
        if (t < E_DIM)
            c = fmaxf(hlast[t], 0.f) * w_edge[E_DIM + t];
        else if (t < E_DIM + G_DIM)
            c = fmaxf(hgraph[t - E_DIM], 0.f) * w_edge[2 * E_DIM + S_DIM + (t - E_DIM)];
        c = wave_sum32(c);
        if (lane == 0) redv[wv] = c;
        __syncthreads();
        if (t < 32) {
            float v = redv[t];
            v = wave_sum32(v);
            if (t == 0) ckS = v + Cseq[k];
        }
        __syncthreads();
        const float Ck = ckS;

        // ---- scores (written before mask update) + first-index argmax
        float bv = -INFINITY; int bi = 0x7fffffff;
        for (int n = t; n < N_POOL; n += 1024) {
            const float sc = maskS[n] * (s0[n] + Ck) + be0;
            scores[(size_t)k * N_POOL + n] = sc;
            if (sc > bv) { bv = sc; bi = n; }          // strict > keeps smallest n per thread
        }
        wave_argmax32(bv, bi);
        if (lane == 0) { redv[wv] = bv; redi[wv] = bi; }
        __syncthreads();
        if (t < 32) {
            bv = redv[t]; bi = redi[t];
            wave_argmax32(bv, bi);
            if (t == 0) { idxS[k] = bi; idxs_out[k] = bi; maskS[bi] = 0.f; }
        }
        __syncthreads();
        const int idx = idxS[k];

        // ---- h_last = Dh[k,k] * (Dh[k,k]*FW[i_k] + Dh[k,k-1]*FW[i_{k-1}])
        if (t < E_DIM) {
            const float ckk = dhc(k, k);
            float q = ckk * FW[(size_t)idx * E_DIM + t];
            if (k >= 1) q += dhc(k - 1, k) * FW[(size_t)idxS[k - 1] * E_DIM + t];
            hlast[t] = ckk * q;
        }

        // ---- h_graph = sum_{j<=k} hg[j]*sigmoid(gg[j]) via shift-stencil on GW/GG rows
        float a0 = 0.f, a1 = 0.f, a2 = 0.f, a3 = 0.f;
        for (int j = jg; j <= k; j += 32) {
            const float cj  = dhc(j, k);
            const int   ij  = idxS[j];
            const float cjm = (j > 0) ? dhc(j - 1, k) : 0.f;
            const int   ijm = (j > 0) ? idxS[j - 1] : 0;
            const float cjp = (j < k) ? dhc(j + 1, k) : 0.f;
            const int   ijp = (j < k) ? idxS[j + 1] : 0;

            const float4 z4 = {0.f, 0.f, 0.f, 0.f};
            const float4 g0 = *(const float4*)(GW + (size_t)ij  * G_DIM + cg);
            const float4 g1 = (j > 0) ? *(const float4*)(GW + (size_t)ijm * G_DIM + cg) : z4;
            const float4 g2 = (j < k) ? *(const float4*)(GW + (size_t)ijp * G_DIM + cg) : z4;
            const float4 e0 = *(const float4*)(GG + (size_t)ij  * G_DIM + cg);
            const float4 e1 = (j > 0) ? *(const float4*)(GG + (size_t)ijm * G_DIM + cg) : z4;
            const float4 e2 = (j < k) ? *(const float4*)(GG + (size_t)ijp * G_DIM + cg) : z4;

            float hg, gg;
            hg = cj * (cj * g0.x + cjm * g1.x + cjp * g2.x) + bg0;
            gg = cj * (cj * e0.x + cjm * e1.x + cjp * e2.x) + bt0;
            a0 += hg * (1.f / (1.f + expf(-gg)));
            hg = cj * (cj * g0.y + cjm * g1.y + cjp * g2.y) + bg1;
            gg = cj * (cj * e0.y + cjm * e1.y + cjp * e2.y) + bt1;
            a1 += hg * (1.f / (1.f + expf(-gg)));
            hg = cj * (cj * g0.z + cjm * g1.z + cjp * g2.z) + bg2;
            gg = cj * (cj * e0.z + cjm * e1.z + cjp * e2.z) + bt2;
            a2 += hg * (1.f / (1.f + expf(-gg)));
            hg = cj * (cj * g0.w + cjm * g1.w + cjp * g2.w) + bg3;
            gg = cj * (cj * e0.w + cjm * e1.w + cjp * e2.w) + bt3;
            a3 += hg * (1.f / (1.f + expf(-gg)));
        }
        part[jg * G_DIM + cg + 0] = a0;
        part[jg * G_DIM + cg + 1] = a1;
        part[jg * G_DIM + cg + 2] = a2;
        part[jg * G_DIM + cg + 3] = a3;
        __syncthreads();
        if (t < G_DIM) {
            float s = 0.f;
            for (int p = 0; p < 32; ++p) s += part[p * G_DIM + t];
            hgraph[t] = s;
        }
        __syncthreads();
    }
}

// ---------------------------------------------------------------------------
extern "C" void kernel_launch(void* const* d_in, const int* in_sizes, int n_in,
                              void* d_out, int out_size, void* d_ws, size_t ws_size,
                              hipStream_t stream)
{
    const float* x       = (const float*)d_in[0];
    const float* f       = (const float*)d_in[1];
    const float* amat    = (const float*)d_in[2];
    const float* dmat    = (const float*)d_in[3];
    const float* w_feat  = (const float*)d_in[4];
    const float* b_feat  = (const float*)d_in[5];
    const float* w_gcn1  = (const float*)d_in[6];
    const float* w_gcn2  = (const float*)d_in[7];
    const float* w_graph = (const float*)d_in[8];
    const float* b_graph = (const float*)d_in[9];
    const float* w_gate  = (const float*)d_in[10];
    const float* b_gate  = (const float*)d_in[11];
    const float* w_edge  = (const float*)d_in[12];
    const float* b_edge  = (const float*)d_in[13];
    const float* w_seq   = (const float*)d_in[14];
    const float* b_seq   = (const float*)d_in[15];

    float* ws  = (float*)d_ws;                 // ~10.6 MB of f32 scratch used
    float* fn  = ws;                           // 4096x64
    float* g   = fn  + N_POOL * E_DIM;         // 4096x64
    float* t1  = g   + N_POOL * E_DIM;         // 4096x64
    float* t2  = t1  + N_POOL * E_DIM;         // 4096x64
    float* t3  = t2  + N_POOL * E_DIM;         // 4096x64
    float* FW  = t3  + N_POOL * E_DIM;         // 4096x64
    float* GW  = FW  + N_POOL * E_DIM;         // 4096x128
    float* GG  = GW  + N_POOL * G_DIM;         // 4096x128
    float* W2G = GG  + N_POOL * G_DIM;         // 64x128
    float* W2T = W2G + E_DIM * G_DIM;          // 64x128
    float* s0  = W2T + E_DIM * G_DIM;          // 4096
    float* Cs  = s0  + N_POOL;                 // 512

    feat_kernel<<<N_POOL, E_DIM, 0, stream>>>(f, w_feat, b_feat, fn, g);
    cseq_kernel<<<2, 256, 0, stream>>>(x, w_seq, b_seq, w_edge, Cs);
    w2_kernel<<<E_DIM, 256, 0, stream>>>(w_gcn2, w_graph, w_gate, W2G, W2T);

    // stripes of 16 rows x 64 cols, 8 waves per 256-thread block
    auto nblk = [](int M, int N) { return ((M / 16) * (N / 64) + 7) / 8; };

    // dAd @ relu(fn) reassociated: t3 = d @ (a @ (d @ g))
    gemm_wmma_f32<<<nblk(N_POOL, E_DIM), 256, 0, stream>>>(dmat, g,  t1, N_POOL, E_DIM, N_POOL, N_POOL, E_DIM, E_DIM);
    gemm_wmma_f32<<<nblk(N_POOL, E_DIM), 256, 0, stream>>>(amat, t1, t2, N_POOL, E_DIM, N_POOL, N_POOL, E_DIM, E_DIM);
    gemm_wmma_f32<<<nblk(N_POOL, E_DIM), 256, 0, stream>>>(dmat, t2, t3, N_POOL, E_DIM, N_POOL, N_POOL, E_DIM, E_DIM);

    post_kernel<<<N_POOL, E_DIM, 0, stream>>>(t3, w_gcn1, w_edge, s0);

    gemm_wmma_f32<<<nblk(N_POOL, E_DIM), 256, 0, stream>>>(fn, w_gcn2, FW, N_POOL, E_DIM, E_DIM, E_DIM, E_DIM, E_DIM);
    gemm_wmma_f32<<<nblk(N_POOL, G_DIM), 256, 0, stream>>>(fn, W2G,    GW, N_POOL, G_DIM, E_DIM, E_DIM, G_DIM, G_DIM);
    gemm_wmma_f32<<<nblk(N_POOL, G_DIM), 256, 0, stream>>>(fn, W2T,    GG, N_POOL, G_DIM, E_DIM, E_DIM, G_DIM, G_DIM);

    float* scoresp  = (float*)d_out;                             // 512x4096 f32
    int*   idxs_out = (int*)(scoresp + (size_t)N_SEQ * N_POOL);  // 512 i32
    scan_kernel<<<1, 1024, 0, stream>>>(s0, Cs, FW, GW, GG, w_edge, b_edge,
                                        b_graph, b_gate, scoresp, idxs_out);
}